// ConnectedThresholdLayer_29884382446232
// MI455X (gfx1250) — compile-verified
//
#include <hip/hip_runtime.h>
#include <stdint.h>

#define DI __device__ __forceinline__

typedef __attribute__((ext_vector_type(4))) float f32x4;
typedef __attribute__((ext_vector_type(4))) int   i32x4;

namespace {
constexpr int   kN    = 1 << 18;        // 262144 nodes per tree
constexpr int   kHW   = 1 << 20;        // 1024*1024 pixels per (b,c)
constexpr int   kBC   = 24;             // B*C
constexpr int   kBCN  = kBC * kN;       // 6,291,456
constexpr int   kTOT  = kBC * kHW;      // 25,165,824
constexpr float kEPS  = 1e-6f;

DI void async_load_b128(unsigned lds_addr, uint64_t gaddr) {
  // CDNA5 async copy: global memory -> LDS, tracked by ASYNCcnt
  asm volatile("global_load_async_to_lds_b128 %0, %1, off"
               :: "v"(lds_addr), "v"(gaddr) : "memory");
}
DI void wait_async_le1() { asm volatile("s_wait_asynccnt 0x1" ::: "memory"); }
DI void wait_async_0()   { asm volatile("s_wait_asynccnt 0x0" ::: "memory"); }

DI float gate_of(float a, float amin, float denom, float s2) {
  float s1   = (a - amin) / denom;
  float soft = 1.0f / (1.0f + expf(s2 - s1));   // sigmoid(s1 - s2)
  return (soft >= 0.5f) ? 1.0f : 0.0f;
}
} // namespace

// ---------------------------------------------------------------- K0: init
__global__ void ctl_init(unsigned* __restrict__ minb, unsigned* __restrict__ maxb) {
  int i = threadIdx.x;
  if (i < 32) { minb[i] = 0x7f7fffffu; maxb[i] = 0u; }   // FLT_MAX bits / 0.0f
}

// ------------------------------------------- K1: per-tree min/max (async LDS)
__global__ __launch_bounds__(256) void ctl_minmax(const float* __restrict__ attr,
                                                  unsigned* __restrict__ minb,
                                                  unsigned* __restrict__ maxb) {
  constexpr int TILES  = 8;
  constexpr int TILE_F = 1024;                     // 256 thr * 4 floats
  __shared__ float buf[2][TILE_F];
  __shared__ float wred[16];
  const int tid  = threadIdx.x;
  const int blk  = blockIdx.x;
  const int tree = blk >> 5;                       // 32 blocks per tree
  const float* gsrc = attr + (size_t)blk * (TILES * TILE_F);
  const unsigned lds0 = (unsigned)(uintptr_t)&buf[0][tid * 4];
  const unsigned lds1 = (unsigned)(uintptr_t)&buf[1][tid * 4];

  async_load_b128(lds0, (uint64_t)(uintptr_t)(gsrc + tid * 4));
  float vmin = 3.402823466e38f, vmax = -3.402823466e38f;
  for (int t = 0; t < TILES; ++t) {
    if (t + 1 < TILES) {
      async_load_b128(((t + 1) & 1) ? lds1 : lds0,
                      (uint64_t)(uintptr_t)(gsrc + (t + 1) * TILE_F + tid * 4));
      wait_async_le1();                            // tile t landed in LDS
    } else {
      wait_async_0();
    }
    __syncthreads();                               // all waves' copies visible
    f32x4 v = *(const f32x4*)&buf[t & 1][tid * 4];
    vmin = fminf(vmin, fminf(fminf(v.x, v.y), fminf(v.z, v.w)));
    vmax = fmaxf(vmax, fmaxf(fmaxf(v.x, v.y), fmaxf(v.z, v.w)));
    __syncthreads();                               // safe to overwrite buffer
  }
  // wave32 reduction, then cross-wave via LDS
  for (int off = 16; off > 0; off >>= 1) {
    vmin = fminf(vmin, __shfl_down(vmin, off));
    vmax = fmaxf(vmax, __shfl_down(vmax, off));
  }
  const int wid = tid >> 5;
  if ((tid & 31) == 0) { wred[wid] = vmin; wred[wid + 8] = vmax; }
  __syncthreads();
  if (tid == 0) {
    float m = wred[0], M = wred[8];
    for (int i = 1; i < 8; ++i) { m = fminf(m, wred[i]); M = fmaxf(M, wred[i + 8]); }
    // attr >= 0, so float ordering == int-bit ordering
    atomicMin((int*)&minb[tree], __float_as_int(m));
    atomicMax((int*)&maxb[tree], __float_as_int(M));
  }
}

// ------------------------------- K2: gate + residue, packed {s, parent} pairs
// Streaming inputs are non-temporal; sp output left RT so it stays L2-resident
// for the chase kernel. The level[] gather stays RT (reused across threads).
__global__ __launch_bounds__(256) void ctl_prep(
    const float* __restrict__ attr, const float* __restrict__ level,
    const float* __restrict__ thr_raw, const int* __restrict__ parent,
    const unsigned* __restrict__ minb, const unsigned* __restrict__ maxb,
    i32x4* __restrict__ sp4) {
  const int t    = blockIdx.x * 256 + threadIdx.x;
  const int e0   = t * 4;
  const int tree = e0 >> 18;
  const int base = tree << 18;
  const float amin  = __uint_as_float(minb[tree]);
  const float amax  = __uint_as_float(maxb[tree]);
  const float denom = fmaxf(amax - amin, kEPS);
  const float s2    = (thr_raw[0] - amin) / denom;
  const f32x4 a = __builtin_nontemporal_load((const f32x4*)(attr + e0));
  const f32x4 l = __builtin_nontemporal_load((const f32x4*)(level + e0));
  const i32x4 p = __builtin_nontemporal_load((const i32x4*)(parent + e0));
  const float* lvl = level + base;
  float sv0 = gate_of(a.x, amin, denom, s2) * (l.x - lvl[p.x]);
  float sv1 = gate_of(a.y, amin, denom, s2) * (l.y - lvl[p.y]);
  float sv2 = gate_of(a.z, amin, denom, s2) * (l.z - lvl[p.z]);
  float sv3 = gate_of(a.w, amin, denom, s2) * (l.w - lvl[p.w]);
  if (e0 == base) sv0 = l.x;                       // root: ungated base level
  i32x4 o0 = { __float_as_int(sv0), p.x, __float_as_int(sv1), p.y };
  i32x4 o1 = { __float_as_int(sv2), p.z, __float_as_int(sv3), p.w };
  sp4[t * 2 + 0] = o0;
  sp4[t * 2 + 1] = o1;
}

// ------------------------------------- K3: parent-chain sum (L2-resident walk)
__global__ __launch_bounds__(256) void ctl_chase(const int2* __restrict__ sp,
                                                 float* __restrict__ S) {
  const int idx  = blockIdx.x * 256 + threadIdx.x;
  const int tree = idx >> 18;
  const int base = tree << 18;
  const int2* tsp = sp + base;
  int  j = idx - base;
  int2 v = tsp[j];
  float acc = __int_as_float(v.x);
  while (j != 0) {                                  // parent[j] < j: terminates
    j = v.y;
    v = tsp[j];
    acc += __int_as_float(v.x);
  }
  S[idx] = acc;                                     // RT: gather kernel reuses S
}

// -------------------------------------------------- K4: pixel gather (float4)
// p2n read and out write are pure streams (200MB) -> non-temporal, so the
// randomly-gathered S (25MB) keeps its L2 residency.
__global__ __launch_bounds__(256) void ctl_gather(const int* __restrict__ p2n,
                                                  const float* __restrict__ S,
                                                  float* __restrict__ out) {
  const int t    = blockIdx.x * 256 + threadIdx.x;
  const int e0   = t * 4;
  const int tree = e0 >> 20;                        // HW = 2^20
  const float* St = S + ((size_t)tree << 18);
  const i32x4 n = __builtin_nontemporal_load((const i32x4*)(p2n + e0));
  f32x4 o = { St[n.x], St[n.y], St[n.z], St[n.w] };
  __builtin_nontemporal_store(o, (f32x4*)(out + e0));
}

// ----------------------- Fallback: fused chase-per-pixel (tiny workspace path)
__global__ __launch_bounds__(256) void ctl_fused(
    const float* __restrict__ attr, const float* __restrict__ level,
    const float* __restrict__ thr_raw, const int* __restrict__ parent,
    const int* __restrict__ p2n,
    const unsigned* __restrict__ minb, const unsigned* __restrict__ maxb,
    float* __restrict__ out) {
  const int idx  = blockIdx.x * 256 + threadIdx.x;
  const int tree = idx >> 20;
  const int base = tree << 18;
  const float amin  = __uint_as_float(minb[tree]);
  const float amax  = __uint_as_float(maxb[tree]);
  const float denom = fmaxf(amax - amin, kEPS);
  const float s2    = (thr_raw[0] - amin) / denom;
  int j = p2n[idx];
  float acc = 0.0f;
  while (true) {
    float lvl = level[base + j];
    if (j == 0) { acc += lvl; break; }
    int par = parent[base + j];
    acc += gate_of(attr[base + j], amin, denom, s2) * (lvl - level[base + par]);
    j = par;
  }
  __builtin_nontemporal_store(acc, out + idx);
}

extern "C" void kernel_launch(void* const* d_in, const int* in_sizes, int n_in,
                              void* d_out, int out_size, void* d_ws, size_t ws_size,
                              hipStream_t stream) {
  (void)in_sizes; (void)n_in; (void)out_size;
  const float*   attr   = (const float*)d_in[0];
  const float*   level  = (const float*)d_in[1];
  const float*   thr    = (const float*)d_in[2];
  const int*     parent = (const int*)d_in[3];
  const int*     p2n    = (const int*)d_in[4];
  float*         out    = (float*)d_out;
  unsigned char* ws     = (unsigned char*)d_ws;

  unsigned* minb = (unsigned*)(ws);
  unsigned* maxb = (unsigned*)(ws + 128);
  i32x4* sp4 = (i32x4*)(ws + 512);
  int2*  sp  = (int2*)(ws + 512);
  float* S   = (float*)(ws + 512 + (size_t)kBCN * 8);
  const size_t need = 512 + (size_t)kBCN * 12;     // packed pairs + S

  ctl_init<<<1, 64, 0, stream>>>(minb, maxb);
  ctl_minmax<<<kBCN / 8192, 256, 0, stream>>>(attr, minb, maxb);

  if (ws_size >= need) {
    ctl_prep <<<kBCN / 1024, 256, 0, stream>>>(attr, level, thr, parent, minb, maxb, sp4);
    ctl_chase<<<kBCN / 256,  256, 0, stream>>>(sp, S);
    ctl_gather<<<kTOT / 1024, 256, 0, stream>>>(p2n, S, out);
  } else {
    ctl_fused<<<kTOT / 256, 256, 0, stream>>>(attr, level, thr, parent, p2n,
                                              minb, maxb, out);
  }
}